// DalleBartDecoderFlax_73615739453958
// MI455X (gfx1250) — compile-verified
//
#include <hip/hip_runtime.h>
#include <math.h>

// ---------------- problem constants (from reference) ----------------
#define E_DIM 1024
#define B_Q   8
#define H_N   16
#define D_H   64
#define G_DIM 4096
#define N_IMG 256
#define T_TXT 64
#define L_NUM 4
#define VOCAB 16385

typedef __attribute__((ext_vector_type(2))) float v2f;
typedef __attribute__((ext_vector_type(8))) float v8f;

// ---------------- block reductions (256 threads) ----------------
__device__ __forceinline__ float block_reduce_sum256(float v, float* red) {
    int tid = threadIdx.x;
    red[tid] = v; __syncthreads();
    for (int off = 128; off > 0; off >>= 1) {
        if (tid < off) red[tid] += red[tid + off];
        __syncthreads();
    }
    float r = red[0]; __syncthreads();
    return r;
}
__device__ __forceinline__ float block_reduce_max256(float v, float* red) {
    int tid = threadIdx.x;
    red[tid] = v; __syncthreads();
    for (int off = 128; off > 0; off >>= 1) {
        if (tid < off) red[tid] = fmaxf(red[tid], red[tid + off]);
        __syncthreads();
    }
    float r = red[0]; __syncthreads();
    return r;
}

// ---------------- WMMA f32 GEMM: C[M,N] = A[M,K] @ W[K,N] (row-major) ---------
// M multiple of 16 (activation rows padded with zeros). One wave per 16-col tile.
// K-loop of V_WMMA_F32_16X16X4_F32 with dual accumulators (halves the dependent
// WMMA chain). Weights streamed with non-temporal loads (read-once, > L2 size);
// activations (A) stay on the regular temporal path (tiny, highly reused).
// A-frag (32-bit A 16x4): lane<16 -> K=0,1 ; lane>=16 -> K=2,3 ; row = lane&15.
// B-frag symmetric; C/D: vgpr r holds row = r + 8*(lane>=16), col = lane&15.
__global__ void gemm_f32_wmma(const float* __restrict__ A, const float* __restrict__ W,
                              float* __restrict__ C, int M, int K, int N,
                              int Mstore, int acc)
{
    const int wave  = threadIdx.x >> 5;
    const int lane  = threadIdx.x & 31;
    const int half  = lane >> 4;        // 0: K+0/1, 1: K+2/3
    const int l15   = lane & 15;
    const int ntile = blockIdx.x * (blockDim.x >> 5) + wave;
    const int col   = ntile * 16 + l15;
    const int mbase = blockIdx.y * 16;
    const int arow  = mbase + l15;
    const bool colok = (col < N);
    const int colc  = colok ? col : (N - 1);   // clamp address, zero value below

    v8f c0 = {0.f,0.f,0.f,0.f,0.f,0.f,0.f,0.f};
    v8f c1 = {0.f,0.f,0.f,0.f,0.f,0.f,0.f,0.f};
    if (acc) {
        #pragma unroll
        for (int r = 0; r < 8; ++r) {
            int crow = mbase + r + 8 * half;
            c0[r] = (colok && crow < Mstore) ? C[(size_t)crow * N + col] : 0.f;
        }
    }

    const float* Arow = A + (size_t)arow * K;
    // K assumed multiple of 8 (1024 or 4096 here)
    for (int k = 0; k < K; k += 8) {
        const int ka = k + half * 2;
        const int kb = k + 4 + half * 2;
        // A fragments: contiguous pair -> single b64 load each
        v2f a0 = *(const v2f*)(Arow + ka);
        v2f a1 = *(const v2f*)(Arow + kb);
        // W fragments: streamed once -> non-temporal loads
        float b0x = __builtin_nontemporal_load(W + (size_t)ka * N + colc);
        float b0y = __builtin_nontemporal_load(W + (size_t)(ka + 1) * N + colc);
        float b1x = __builtin_nontemporal_load(W + (size_t)kb * N + colc);
        float b1y = __builtin_nontemporal_load(W + (size_t)(kb + 1) * N + colc);
        v2f b0, b1;
        b0.x = colok ? b0x : 0.f;
        b0.y = colok ? b0y : 0.f;
        b1.x = colok ? b1x : 0.f;
        b1.y = colok ? b1y : 0.f;
        c0 = __builtin_amdgcn_wmma_f32_16x16x4_f32(false, a0, false, b0,
                                                   (short)0, c0, false, false);
        c1 = __builtin_amdgcn_wmma_f32_16x16x4_f32(false, a1, false, b1,
                                                   (short)0, c1, false, false);
    }

    if (colok) {
        #pragma unroll
        for (int r = 0; r < 8; ++r) {
            int crow = mbase + r + 8 * half;
            if (crow < Mstore) C[(size_t)crow * N + col] = c0[r] + c1[r];
        }
    }
}

// ---------------- embedding + LayerNorm (scale+bias) ----------------
__global__ void embed_ln_kernel(const int* __restrict__ ptok, const int* __restrict__ tix,
                                const float* __restrict__ etok, const float* __restrict__ epos,
                                const float* __restrict__ s, const float* __restrict__ b,
                                float* __restrict__ out)
{
    __shared__ float red[256];
    const int row = blockIdx.x, tid = threadIdx.x;
    const size_t tb = (size_t)(*ptok) * E_DIM;
    const size_t pb = (size_t)(*tix) * E_DIM;
    float v[4]; float sum = 0.f;
    #pragma unroll
    for (int i = 0; i < 4; ++i) {
        int idx = tid + i * 256;
        v[i] = etok[tb + idx] + epos[pb + idx];
        sum += v[i];
    }
    float mean = block_reduce_sum256(sum, red) * (1.f / E_DIM);
    float sq = 0.f;
    #pragma unroll
    for (int i = 0; i < 4; ++i) { float d = v[i] - mean; sq += d * d; }
    float var = block_reduce_sum256(sq, red) * (1.f / E_DIM);
    float inv = rsqrtf(var + 1e-6f);
    #pragma unroll
    for (int i = 0; i < 4; ++i) {
        int idx = tid + i * 256;
        out[(size_t)row * E_DIM + idx] = (v[i] - mean) * inv * s[idx] + b[idx];
    }
}

// ---------------- LayerNorm over E (optional scale), out = LN(in) -------------
__global__ void ln_row_kernel(const float* __restrict__ in, const float* __restrict__ bias,
                              const float* __restrict__ scale, float* __restrict__ out)
{
    __shared__ float red[256];
    const int row = blockIdx.x, tid = threadIdx.x;
    float v[4]; float sum = 0.f;
    #pragma unroll
    for (int i = 0; i < 4; ++i) { v[i] = in[(size_t)row * E_DIM + tid + i * 256]; sum += v[i]; }
    float mean = block_reduce_sum256(sum, red) * (1.f / E_DIM);
    float sq = 0.f;
    #pragma unroll
    for (int i = 0; i < 4; ++i) { float d = v[i] - mean; sq += d * d; }
    float var = block_reduce_sum256(sq, red) * (1.f / E_DIM);
    float inv = rsqrtf(var + 1e-6f);
    #pragma unroll
    for (int i = 0; i < 4; ++i) {
        int idx = tid + i * 256;
        float y = (v[i] - mean) * inv;
        if (scale) y *= scale[idx];
        out[(size_t)row * E_DIM + idx] = y + bias[idx];
    }
}

// ---------------- x += LN(h, bias, scale) ----------------
__global__ void ln_residual_kernel(const float* __restrict__ h, const float* __restrict__ bias,
                                   const float* __restrict__ scale, float* __restrict__ x)
{
    __shared__ float red[256];
    const int row = blockIdx.x, tid = threadIdx.x;
    float v[4]; float sum = 0.f;
    #pragma unroll
    for (int i = 0; i < 4; ++i) { v[i] = h[(size_t)row * E_DIM + tid + i * 256]; sum += v[i]; }
    float mean = block_reduce_sum256(sum, red) * (1.f / E_DIM);
    float sq = 0.f;
    #pragma unroll
    for (int i = 0; i < 4; ++i) { float d = v[i] - mean; sq += d * d; }
    float var = block_reduce_sum256(sq, red) * (1.f / E_DIM);
    float inv = rsqrtf(var + 1e-6f);
    #pragma unroll
    for (int i = 0; i < 4; ++i) {
        int idx = tid + i * 256;
        x[(size_t)row * E_DIM + idx] += (v[i] - mean) * inv * scale[idx] + bias[idx];
    }
}

// ---------------- write k,v rows into cache copy at token_index ----------------
__global__ void cache_update_kernel(const float* __restrict__ kb, const float* __restrict__ vb,
                                    float* __restrict__ astL, const int* __restrict__ tix)
{
    const int r = blockIdx.x;                 // 0..15 : [0..7]=k rows, [8..15]=v rows
    const int tid = threadIdx.x;
    const int t = *tix;
    const float* src = (r < B_Q) ? (kb + (size_t)r * E_DIM) : (vb + (size_t)(r - B_Q) * E_DIM);
    float* dst = astL + ((size_t)r * N_IMG + t) * E_DIM;
    #pragma unroll
    for (int i = 0; i < 4; ++i) dst[tid + i * 256] = src[tid + i * 256];
}

// ---------------- self attention: Tk = 256, causal mask j <= token_index -------
__global__ void self_attn_kernel(const float* __restrict__ qb, const float* __restrict__ astL,
                                 float* __restrict__ out, const int* __restrict__ tix)
{
    __shared__ float p[N_IMG];
    __shared__ float red[256];
    const int b = blockIdx.x, h = blockIdx.y, tid = threadIdx.x;
    const int valid = *tix + 1;
    const float* q  = qb + (size_t)b * E_DIM + h * D_H;
    const float* Kb = astL + (size_t)b * N_IMG * E_DIM;
    const float* Vb = astL + (size_t)(B_Q + b) * N_IMG * E_DIM;
    float s = -3.0e38f;
    if (tid < valid) {
        const float* kr = Kb + (size_t)tid * E_DIM + h * D_H;
        float acc = 0.f;
        for (int d = 0; d < D_H; ++d) acc = fmaf(q[d], kr[d], acc);
        s = acc * 0.125f;   // 1/sqrt(64)
    }
    float mx = block_reduce_max256(s, red);
    float e = (tid < valid) ? __expf(s - mx) : 0.f;
    p[tid] = e;
    float sum = block_reduce_sum256(e, red);
    float inv = 1.f / sum;
    __syncthreads();
    if (tid < D_H) {
        float acc = 0.f;
        for (int j = 0; j < valid; ++j)
            acc = fmaf(p[j], Vb[(size_t)j * E_DIM + h * D_H + tid], acc);
        out[(size_t)b * E_DIM + h * D_H + tid] = acc * inv;
    }
}

// ---------------- cross attention: Tk = 64, mask from attention_mask -----------
__global__ void cross_attn_kernel(const float* __restrict__ qb, const float* __restrict__ enck,
                                  const float* __restrict__ encv,
                                  const unsigned char* __restrict__ mask, float* __restrict__ out)
{
    __shared__ float p[T_TXT];
    __shared__ float red[64];
    const int b = blockIdx.x, h = blockIdx.y, tid = threadIdx.x; // 64 threads
    const float* q  = qb + (size_t)b * E_DIM + h * D_H;
    const float* Kb = enck + (size_t)b * T_TXT * E_DIM;
    const float* Vb = encv + (size_t)b * T_TXT * E_DIM;
    const bool mk = mask[b * T_TXT + tid] != 0;
    float acc = 0.f;
    const float* kr = Kb + (size_t)tid * E_DIM + h * D_H;
    for (int d = 0; d < D_H; ++d) acc = fmaf(q[d], kr[d], acc);
    float s = mk ? acc * 0.125f : -3.0e38f;
    red[tid] = s; __syncthreads();
    for (int off = 32; off > 0; off >>= 1) { if (tid < off) red[tid] = fmaxf(red[tid], red[tid + off]); __syncthreads(); }
    float mx = red[0]; __syncthreads();
    float e = mk ? __expf(s - mx) : 0.f;
    p[tid] = e; red[tid] = e; __syncthreads();
    for (int off = 32; off > 0; off >>= 1) { if (tid < off) red[tid] += red[tid + off]; __syncthreads(); }
    float inv = 1.f / red[0]; __syncthreads();
    float o = 0.f;
    for (int j = 0; j < T_TXT; ++j)
        o = fmaf(p[j], Vb[(size_t)j * E_DIM + h * D_H + tid], o);
    out[(size_t)b * E_DIM + h * D_H + tid] = o * inv;
}

// ---------------- GLU gate: LN( gelu_exact(t0) * t1 , bias ) over G=4096 -------
__global__ void glu_ln_kernel(const float* __restrict__ t0, const float* __restrict__ t1,
                              const float* __restrict__ bias, float* __restrict__ out)
{
    __shared__ float red[256];
    const int row = blockIdx.x, tid = threadIdx.x;
    float u[16]; float s = 0.f, sq = 0.f;
    #pragma unroll
    for (int i = 0; i < 16; ++i) {
        int idx = tid + i * 256;
        float a = t0[(size_t)row * G_DIM + idx];
        float g = 0.5f * a * (1.f + erff(a * 0.70710678118654752f)); // exact gelu
        float val = g * t1[(size_t)row * G_DIM + idx];
        u[i] = val; s += val; sq += val * val;
    }
    float mean = block_reduce_sum256(s, red) * (1.f / G_DIM);
    float ms   = block_reduce_sum256(sq, red) * (1.f / G_DIM);
    float inv  = rsqrtf(ms - mean * mean + 1e-6f);
    #pragma unroll
    for (int i = 0; i < 16; ++i) {
        int idx = tid + i * 256;
        out[(size_t)row * G_DIM + idx] = (u[i] - mean) * inv + bias[idx];
    }
}

// =====================================================================
extern "C" void kernel_launch(void* const* d_in, const int* in_sizes, int n_in,
                              void* d_out, int out_size, void* d_ws, size_t ws_size,
                              hipStream_t stream)
{
    const float*         encoder_state = (const float*)d_in[0];
    const float*         ast_in        = (const float*)d_in[1];
    const unsigned char* attn_mask     = (const unsigned char*)d_in[2];
    const int*           prev_tok      = (const int*)d_in[3];
    const int*           tok_idx       = (const int*)d_in[4];
    const float* embed_tok   = (const float*)d_in[5];
    const float* embed_pos   = (const float*)d_in[6];
    const float* lnemb_s     = (const float*)d_in[7];
    const float* lnemb_b     = (const float*)d_in[8];
    const float* pre_self_b  = (const float*)d_in[9];
    const float* self_q      = (const float*)d_in[10];
    const float* self_k      = (const float*)d_in[11];
    const float* self_v      = (const float*)d_in[12];
    const float* self_o      = (const float*)d_in[13];
    const float* self_ln_s   = (const float*)d_in[14];
    const float* self_ln_b   = (const float*)d_in[15];
    const float* pre_enc_b   = (const float*)d_in[16];
    const float* enc_q       = (const float*)d_in[17];
    const float* enc_k       = (const float*)d_in[18];
    const float* enc_v       = (const float*)d_in[19];
    const float* enc_o       = (const float*)d_in[20];
    const float* enc_ln_s    = (const float*)d_in[21];
    const float* enc_ln_b    = (const float*)d_in[22];
    const float* glu_ln0_b   = (const float*)d_in[23];
    const float* glu_fc0     = (const float*)d_in[24];
    const float* glu_fc1     = (const float*)d_in[25];
    const float* glu_ln1_b   = (const float*)d_in[26];
    const float* glu_fc2     = (const float*)d_in[27];
    const float* final_ln_b  = (const float*)d_in[28];
    const float* lm_head     = (const float*)d_in[29];

    float* logits  = (float*)d_out;
    float* ast_out = logits + (size_t)B_Q * VOCAB;   // [L, 2B, N_IMG, E]

    // ---- scratch layout (activations padded to 16 rows; pad rows zero) ----
    float* xpad  = (float*)d_ws;                 // [16,E]
    float* hpad  = xpad  + 16 * E_DIM;           // [16,E]
    float* qpad  = hpad  + 16 * E_DIM;           // [16,E]
    float* kpad  = qpad  + 16 * E_DIM;           // [16,E]
    float* vpad  = kpad  + 16 * E_DIM;           // [16,E]
    float* attnp = vpad  + 16 * E_DIM;           // [16,E]
    float* t0    = attnp + 16 * E_DIM;           // [16,G]
    float* t1    = t0    + 16 * G_DIM;           // [16,G]
    float* gpad  = t1    + 16 * G_DIM;           // [16,G]
    float* enckb = gpad  + 16 * G_DIM;           // [B*T_TXT, E]
    float* encvb = enckb + (size_t)B_Q * T_TXT * E_DIM;
    const size_t ws_floats = 6 * 16 * E_DIM + 3 * 16 * G_DIM + 2 * (size_t)B_Q * T_TXT * E_DIM;

    hipMemsetAsync(d_ws, 0, ws_floats * sizeof(float), stream);
    hipMemcpyAsync(ast_out, ast_in,
                   (size_t)L_NUM * 2 * B_Q * N_IMG * E_DIM * sizeof(float),
                   hipMemcpyDeviceToDevice, stream);

    embed_ln_kernel<<<B_Q, 256, 0, stream>>>(prev_tok, tok_idx, embed_tok, embed_pos,
                                             lnemb_s, lnemb_b, xpad);

    auto gemm = [&](const float* A, const float* W, float* C,
                    int M, int K, int N, int Mstore, int acc) {
        int ntiles = (N + 15) / 16;
        dim3 grid((ntiles + 7) / 8, M / 16, 1);   // 8 waves (256 thr) per block
        gemm_f32_wmma<<<grid, 256, 0, stream>>>(A, W, C, M, K, N, Mstore, acc);
    };

    for (int l = 0; l < L_NUM; ++l) {
        float* astL = ast_out + (size_t)l * 2 * B_Q * N_IMG * E_DIM;
        const size_t wEE = (size_t)l * E_DIM * E_DIM;
        const size_t wEG = (size_t)l * E_DIM * G_DIM;

        // ---- self attention block ----
        ln_row_kernel<<<B_Q, 256, 0, stream>>>(xpad, pre_self_b + l * E_DIM, nullptr, hpad);
        gemm(hpad, self_q + wEE, qpad, 16, E_DIM, E_DIM, B_Q, 0);
        gemm(hpad, self_k + wEE, kpad, 16, E_DIM, E_DIM, B_Q, 0);
        gemm(hpad, self_v + wEE, vpad, 16, E_DIM, E_DIM, B_Q, 0);
        cache_update_kernel<<<2 * B_Q, 256, 0, stream>>>(kpad, vpad, astL, tok_idx);
        self_attn_kernel<<<dim3(B_Q, H_N), 256, 0, stream>>>(qpad, astL, attnp, tok_idx);
        gemm(attnp, self_o + wEE, hpad, 16, E_DIM, E_DIM, B_Q, 0);
        ln_residual_kernel<<<B_Q, 256, 0, stream>>>(hpad, self_ln_b + l * E_DIM,
                                                    self_ln_s + l * E_DIM, xpad);

        // ---- cross attention block ----
        ln_row_kernel<<<B_Q, 256, 0, stream>>>(xpad, pre_enc_b + l * E_DIM, nullptr, hpad);
        gemm(encoder_state, enc_k + wEE, enckb, B_Q * T_TXT, E_DIM, E_DIM, B_Q * T_TXT, 0);
        gemm(encoder_state, enc_v + wEE, encvb, B_Q * T_TXT, E_DIM, E_DIM, B_Q * T_TXT, 0);
        gemm(hpad, enc_q + wEE, qpad, 16, E_DIM, E_DIM, B_Q, 0);
        cross_attn_kernel<<<dim3(B_Q, H_N), 64, 0, stream>>>(qpad, enckb, encvb, attn_mask, attnp);
        gemm(attnp, enc_o + wEE, hpad, 16, E_DIM, E_DIM, B_Q, 0);
        ln_residual_kernel<<<B_Q, 256, 0, stream>>>(hpad, enc_ln_b + l * E_DIM,
                                                    enc_ln_s + l * E_DIM, xpad);

        // ---- GLU block ----
        ln_row_kernel<<<B_Q, 256, 0, stream>>>(xpad, glu_ln0_b + l * E_DIM, nullptr, hpad);
        gemm(hpad, glu_fc0 + wEG, t0, 16, E_DIM, G_DIM, B_Q, 0);
        gemm(hpad, glu_fc1 + wEG, t1, 16, E_DIM, G_DIM, B_Q, 0);
        glu_ln_kernel<<<B_Q, 256, 0, stream>>>(t0, t1, glu_ln1_b + l * G_DIM, gpad);
        gemm(gpad, glu_fc2 + (size_t)l * G_DIM * E_DIM, xpad, 16, G_DIM, E_DIM, B_Q, 1);
    }

    // ---- head ----
    ln_row_kernel<<<B_Q, 256, 0, stream>>>(xpad, final_ln_b, nullptr, hpad);
    gemm(hpad, lm_head, logits, 16, E_DIM, VOCAB, B_Q, 0);
}